// PositiveProb_53111565582670
// MI455X (gfx1250) — compile-verified
//
#include <hip/hip_runtime.h>
#include <hip/hip_bf16.h>

typedef __attribute__((ext_vector_type(2))) float v2f;
typedef __attribute__((ext_vector_type(8))) float v8f;

#define NTOTAL   (512 * 2 * 256)     // 262144 diagonal contributions
#define NBLOCKS  32
#define NTHREADS 256
#define NWAVES   (NTHREADS / 32)
#define PER_THREAD (NTOTAL / (NBLOCKS * NTHREADS))   // 32

// Wave32 full reduction of per-lane float p using V_WMMA_F32_16X16X4_F32.
// A (16x4 f32): vgpr0 = p (lanes 0-15 -> K=0, lanes 16-31 -> K=2), vgpr1 = 0.
// B (4x16 f32): all ones  ->  D[m][n] = p_m + p_{m+16} (replicated over n).
// Lane L<16 holds D rows 0..7 in its 8 acc VGPRs; lane L>=16 holds rows 8..15.
// Sum the 8 VGPRs, then shfl_xor(16) to combine the two half-sums.
__device__ __forceinline__ float wave_reduce_wmma_f32(float p) {
    v2f a; a.x = p;    a.y = 0.0f;
    v2f b; b.x = 1.0f; b.y = 1.0f;
    v8f c = {};
    v8f d = __builtin_amdgcn_wmma_f32_16x16x4_f32(
        /*neg_a=*/false, a, /*neg_b=*/false, b,
        /*c_mod=*/(short)0, c, /*reuse_a=*/false, /*reuse_b=*/false);
    float s = d[0] + d[1] + d[2] + d[3] + d[4] + d[5] + d[6] + d[7];
    s += __shfl_xor(s, 16, 32);
    return s;   // full wave sum, in every lane
}

__global__ __launch_bounds__(NTHREADS)
void pp_partial_kernel(const float* __restrict__ z1,
                       const float* __restrict__ z2,
                       float* __restrict__ partials) {
    const int tid    = blockIdx.x * NTHREADS + threadIdx.x;
    const int stride = NBLOCKS * NTHREADS;

    float p = 0.0f;
#pragma unroll 4
    for (int it = 0; it < PER_THREAD; ++it) {
        const int idx = tid + it * stride;          // flat (i, k, m)
        const int m = idx & 255;
        const int k = (idx >> 8) & 1;
        const int i = idx >> 9;
        // z1 row i: [0:256) = means, [256:512) = logvars
        const float mean = z1[(i << 9) + m];
        const float lv   = z1[(i << 9) + 256 + m];
        // z2 reshaped (K, B, M): element (k, i, m)
        const float x    = z2[(k << 17) + (i << 8) + m];
        const float inv_std = __expf(-0.5f * lv);
        const float dd      = (x - mean) * inv_std;
        // exp(-0.5*d^2 - log(std)) ; the 1/sqrt(2*pi) factor is applied once at the end
        p += __expf(-0.5f * (dd * dd + lv));
    }

    // full wave active here (no divergence above): EXEC all-1s for WMMA
    const float s = wave_reduce_wmma_f32(p);

    __shared__ float wsum[NWAVES];
    const int lane = threadIdx.x & 31;
    const int wave = threadIdx.x >> 5;
    if (lane == 0) wsum[wave] = s;
    __syncthreads();

    if (threadIdx.x == 0) {
        float t = 0.0f;
#pragma unroll
        for (int w = 0; w < NWAVES; ++w) t += wsum[w];
        partials[blockIdx.x] = t;
    }
}

__global__ void pp_finalize_kernel(const float* __restrict__ partials,
                                   float* __restrict__ out) {
    if (threadIdx.x == 0 && blockIdx.x == 0) {
        float s = 0.0f;
#pragma unroll
        for (int b = 0; b < NBLOCKS; ++b) s += partials[b];   // fixed order: deterministic
        // scale: 1/sqrt(2*pi) (factored-out constant) * 1/(B*K*M)  with B*K*M == B*B
        out[0] = s * (0.3989422804014327f / (float)NTOTAL);
    }
}

extern "C" void kernel_launch(void* const* d_in, const int* in_sizes, int n_in,
                              void* d_out, int out_size, void* d_ws, size_t ws_size,
                              hipStream_t stream) {
    const float* z1 = (const float*)d_in[0];   // (512, 512)  = (B, 2M)
    const float* z2 = (const float*)d_in[1];   // (1024, 256) = (K*B, M)
    float* out      = (float*)d_out;           // scalar
    float* partials = (float*)d_ws;            // 32 floats of scratch

    pp_partial_kernel<<<NBLOCKS, NTHREADS, 0, stream>>>(z1, z2, partials);
    pp_finalize_kernel<<<1, 32, 0, stream>>>(partials, out);
}